// DistanceWeightedSelfAttention_84327387890254
// MI455X (gfx1250) — compile-verified
//
#include <hip/hip_runtime.h>
#include <hip/hip_bf16.h>

#define D_MODEL 256
#define SEQ     4096
#define NBATCH  8
#define LAMBDA  0.5f
#define NEG_INF_F (-1000000000.0f)

#define ROWS_PER_WAVE  16
#define WAVES_PER_BLK  8
#define ROWS_PER_BLK   (ROWS_PER_WAVE * WAVES_PER_BLK)   // 128
#define BLOCK          256

typedef __attribute__((ext_vector_type(2))) float v2f;
typedef __attribute__((ext_vector_type(8))) float v8f;

__global__ __launch_bounds__(BLOCK)
void dwsa_kernel(const float* __restrict__ x,
                 const float* __restrict__ Wq,
                 const float* __restrict__ Wk,
                 const float* __restrict__ Wv,
                 float* __restrict__ out)
{
    __shared__ float2 zh[SEQ];                     // (size, height) per key, 32 KB
    __shared__ float  red[BLOCK];                  // dot(Wq,Wk) reduction
    __shared__ float  wv_s[D_MODEL];               // Wv * 0.25 (K=4 replication)
    __shared__ float  w_lds[WAVES_PER_BLK][ROWS_PER_WAVE];

    const int tid  = threadIdx.x;
    const int lane = tid & 31;
    const int wave = tid >> 5;

    const int blocks_per_b = SEQ / ROWS_PER_BLK;   // 32
    const int b      = blockIdx.x / blocks_per_b;
    const int rb     = blockIdx.x % blocks_per_b;
    const int s_wave = rb * ROWS_PER_BLK + wave * ROWS_PER_WAVE;

    // ---- stage x[b,:,:] into LDS (float2 = (size, height)) ----
    const float2* xb = reinterpret_cast<const float2*>(x) + (size_t)b * SEQ;
    for (int i = tid; i < SEQ; i += BLOCK) zh[i] = xb[i];

    // ---- stage Wv (pre-scaled by 1/4 for the K=4 replicated WMMA) ----
    wv_s[tid] = 0.25f * Wv[tid];

    // ---- scalar c = dot(Wq,Wk) / sqrt(D_MODEL) ----
    red[tid] = Wq[tid] * Wk[tid];
    __syncthreads();
    #pragma unroll
    for (int off = BLOCK / 2; off > 0; off >>= 1) {
        if (tid < off) red[tid] += red[tid + off];
        __syncthreads();
    }
    const float cscale = red[0] * (1.0f / 16.0f);

    // ---- per-row query constants (16 rows per wave, register blocked) ----
    float zs[ROWS_PER_WAVE], aa[ROWS_PER_WAVE];
    #pragma unroll
    for (int r = 0; r < ROWS_PER_WAVE; ++r) {
        float2 q = zh[s_wave + r];
        zs[r] = q.x;                 // size_s
        aa[r] = cscale * q.y;        // c * h_s
    }

    // ---- pass 1: exact row maxima over all keys ----
    float m[ROWS_PER_WAVE];
    #pragma unroll
    for (int r = 0; r < ROWS_PER_WAVE; ++r) m[r] = -3.0e38f;

    for (int t = lane; t < SEQ; t += 32) {
        float2 kt = zh[t];
        float  mh = (kt.y == 0.0f) ? NEG_INF_F : 0.0f;   // key mask
        #pragma unroll
        for (int r = 0; r < ROWS_PER_WAVE; ++r) {
            float l = fmaf(aa[r], kt.y, mh - LAMBDA * fabsf(zs[r] - kt.x));
            m[r] = fmaxf(m[r], l);
        }
    }
    #pragma unroll
    for (int r = 0; r < ROWS_PER_WAVE; ++r) {
        #pragma unroll
        for (int off = 16; off > 0; off >>= 1)
            m[r] = fmaxf(m[r], __shfl_xor(m[r], off, 32));
    }

    // ---- pass 2: softmax denominator and h-weighted numerator ----
    float dsum[ROWS_PER_WAVE], wsum[ROWS_PER_WAVE];
    #pragma unroll
    for (int r = 0; r < ROWS_PER_WAVE; ++r) { dsum[r] = 0.0f; wsum[r] = 0.0f; }

    for (int t = lane; t < SEQ; t += 32) {
        float2 kt = zh[t];
        float  mh = (kt.y == 0.0f) ? NEG_INF_F : 0.0f;
        #pragma unroll
        for (int r = 0; r < ROWS_PER_WAVE; ++r) {
            float l = fmaf(aa[r], kt.y, mh - LAMBDA * fabsf(zs[r] - kt.x));
            float e = __expf(l - m[r]);          // masked keys underflow to 0
            dsum[r] += e;
            wsum[r]  = fmaf(e, kt.y, wsum[r]);   // h_t == 0 for masked keys
        }
    }
    #pragma unroll
    for (int r = 0; r < ROWS_PER_WAVE; ++r) {
        #pragma unroll
        for (int off = 16; off > 0; off >>= 1) {
            dsum[r] += __shfl_xor(dsum[r], off, 32);
            wsum[r] += __shfl_xor(wsum[r], off, 32);
        }
    }

    // w[row] = numerator / denominator  (denominator >= 1 since max is exact)
    if (lane == 0) {
        #pragma unroll
        for (int r = 0; r < ROWS_PER_WAVE; ++r)
            w_lds[wave][r] = wsum[r] / dsum[r];
    }
    __syncthreads();

    // ---- output: out[s, :] = w[s] * Wv  via V_WMMA_F32_16X16X4_F32 ----
    // A[m,k] = w[m] for all k; B[k,n] = 0.25*Wv[n] for all k  =>  D = w ⊗ Wv,
    // independent of the exact K-slot striping. EXEC is all-1s here.
    const float aval = w_lds[wave][lane & 15];
    v2f A;  A[0] = aval;  A[1] = aval;
    v8f C = {0.f, 0.f, 0.f, 0.f, 0.f, 0.f, 0.f, 0.f};

    const int    nloc  = lane & 15;
    const size_t obase = (size_t)(b * SEQ + s_wave + ((lane < 16) ? 0 : 8)) * D_MODEL;

    #pragma unroll
    for (int tile = 0; tile < D_MODEL / 16; ++tile) {
        float bval = wv_s[tile * 16 + nloc];
        v2f Bv;  Bv[0] = bval;  Bv[1] = bval;
        v8f D = __builtin_amdgcn_wmma_f32_16x16x4_f32(
            /*neg_a=*/false, A, /*neg_b=*/false, Bv,
            /*c_mod=*/(short)0, C, /*reuse_a=*/false, /*reuse_b=*/false);
        float* op = out + obase + (size_t)(tile * 16 + nloc);
        #pragma unroll
        for (int j = 0; j < 8; ++j)
            op[(size_t)j * D_MODEL] = D[j];      // D vgpr j -> row (j or j+8)
    }
}

extern "C" void kernel_launch(void* const* d_in, const int* in_sizes, int n_in,
                              void* d_out, int out_size, void* d_ws, size_t ws_size,
                              hipStream_t stream)
{
    const float* x  = (const float*)d_in[0];
    const float* Wq = (const float*)d_in[1];
    const float* Wk = (const float*)d_in[2];
    const float* Wv = (const float*)d_in[3];
    float* out = (float*)d_out;

    dim3 grid(NBATCH * (SEQ / ROWS_PER_BLK));    // 8 * 32 = 256 blocks
    dim3 block(BLOCK);
    dwsa_kernel<<<grid, block, 0, stream>>>(x, Wq, Wk, Wv, out);
}